// CapsuleLayer_61426622267775
// MI455X (gfx1250) — compile-verified
//
#include <hip/hip_runtime.h>

typedef __attribute__((ext_vector_type(2))) float v2f;
typedef __attribute__((ext_vector_type(8))) float v8f;

#define B_SZ 256
#define N_SZ 1152
#define C_SZ 10
#define O_SZ 16
#define I_SZ 8
#define KO   160    // C*OUT  (columns of s / rows of Y)
#define NI   9216   // N*IN   (contraction dim of s-GEMM / columns of Y)

// ---------------- small elementwise kernels ----------------

__global__ void zero_kernel(float* __restrict__ p, int n) {
  int i = blockIdx.x * blockDim.x + threadIdx.x;
  if (i < n) p[i] = 0.0f;
}

// c[n,:] = softmax(b[n,:]) over C_SZ=10 capsules
__global__ void softmax_kernel(const float* __restrict__ b, float* __restrict__ c) {
  int n = blockIdx.x * blockDim.x + threadIdx.x;
  if (n >= N_SZ) return;
  float row[C_SZ];
  float m = -3.402823e38f;
  #pragma unroll
  for (int k = 0; k < C_SZ; ++k) { row[k] = b[n * C_SZ + k]; m = fmaxf(m, row[k]); }
  float s = 0.0f;
  #pragma unroll
  for (int k = 0; k < C_SZ; ++k) { row[k] = __expf(row[k] - m); s += row[k]; }
  float inv = 1.0f / s;
  #pragma unroll
  for (int k = 0; k < C_SZ; ++k) c[n * C_SZ + k] = row[k] * inv;
}

// Wct[(n*8+i), (k*16+o)] = c[n,k] * W[n,k,o,i]   (K-major layout for s-GEMM B operand)
__global__ void make_wct_kernel(const float* __restrict__ W,
                                const float* __restrict__ c,
                                float* __restrict__ Wct) {
  int t = blockIdx.x * blockDim.x + threadIdx.x;
  if (t >= NI * KO) return;
  int row = t / KO;          // n*8 + i
  int col = t - row * KO;    // k*16 + o
  int n = row >> 3, i = row & 7;
  int k = col >> 4, o = col & 15;
  Wct[t] = c[n * C_SZ + k] * W[((n * C_SZ + k) * O_SZ + o) * I_SZ + i];
}

// v = s*|s|/(1+s^2); optionally mirror into d_out on the last iteration
__global__ void squash_kernel(const float* __restrict__ S,
                              float* __restrict__ V,
                              float* __restrict__ Out) {
  int t = blockIdx.x * blockDim.x + threadIdx.x;
  if (t >= B_SZ * KO) return;
  float s = S[t];
  float v = s * fabsf(s) / (1.0f + s * s);
  V[t] = v;
  if (Out) Out[t] = v;
}

// b[n,k] += (1/B) * sum_{o,i} W[n,k,o,i] * Y[(k*16+o), (n*8+i)]
__global__ void update_b_kernel(const float* __restrict__ W,
                                const float* __restrict__ Y,
                                float* __restrict__ bij) {
  int t = blockIdx.x * blockDim.x + threadIdx.x;
  if (t >= N_SZ * C_SZ) return;
  int n = t / C_SZ, k = t - n * C_SZ;
  const float* w = W + (n * C_SZ + k) * O_SZ * I_SZ;   // 128 contiguous floats [o][i]
  float dot = 0.0f;
  #pragma unroll
  for (int o = 0; o < O_SZ; ++o)
    #pragma unroll
    for (int i = 0; i < I_SZ; ++i)
      dot = fmaf(w[o * I_SZ + i], Y[(k * O_SZ + o) * NI + n * I_SZ + i], dot);
  bij[t] += dot * (1.0f / (float)B_SZ);
}

// ---------------- WMMA f32 16x16x4 GEMMs ----------------
// Fragment layouts (ISA 7.12.2, wave32):
//   A 16x4:  lane<16 -> M=lane,    reg r holds K = r + 2*(lane>=16)
//   B 4x16:  lane<16 -> N=lane&15, reg r holds K = r + 2*(lane>=16)
//   C/D 16x16: reg r holds M = r + 8*(lane>=16), N = lane&15

// S[256,160] = X[256,9216] @ Wct[9216,160]; one 16x16 tile per wave, full-K serial.
__global__ void __launch_bounds__(256) gemm_s_kernel(const float* __restrict__ X,
                                                     const float* __restrict__ Wct,
                                                     float* __restrict__ S) {
  int wave = threadIdx.x >> 5;
  int lane = threadIdx.x & 31;
  int tile = blockIdx.x * 8 + wave;            // 0..159
  int mt = tile / 10, nt = tile - mt * 10;
  int m0 = mt * 16, n0 = nt * 16;
  int half = lane >> 4, lr = lane & 15;

  const float* arow = X + (m0 + lr) * NI + 2 * half;     // A[m0+lr, k0+2h+{0,1}]
  const float* bcol = Wct + 2 * half * KO + n0 + lr;     // B[k0+2h+{0,1}, n0+lr]

  v8f acc = {};
  #pragma unroll 4
  for (int k0 = 0; k0 < NI; k0 += 4) {
    v2f a, b;
    a.x = arow[k0];
    a.y = arow[k0 + 1];
    b.x = bcol[k0 * KO];
    b.y = bcol[k0 * KO + KO];
    acc = __builtin_amdgcn_wmma_f32_16x16x4_f32(false, a, false, b,
                                                (short)0, acc, false, false);
  }

  float* srow = S + (m0 + 8 * half) * KO + n0 + lr;
  #pragma unroll
  for (int r = 0; r < 8; ++r) srow[r * KO] = acc[r];
}

// Y[160,9216] = V^T[160,256] @ X[256,9216]; 5760 tiles, K=256.
__global__ void __launch_bounds__(256) gemm_y_kernel(const float* __restrict__ V,
                                                     const float* __restrict__ X,
                                                     float* __restrict__ Y) {
  int wave = threadIdx.x >> 5;
  int lane = threadIdx.x & 31;
  int tile = blockIdx.x * 8 + wave;            // 0..5759
  int mt = tile / 576, nt = tile - mt * 576;
  int m0 = mt * 16, n0 = nt * 16;
  int half = lane >> 4, lr = lane & 15;

  // A[m,k] = V[k,m]  -> V[(k0+2h+j)*160 + m0+lr]
  const float* acol = V + 2 * half * KO + m0 + lr;
  // B[k,n] = X[k,n]  -> X[(k0+2h+j)*9216 + n0+lr]
  const float* bcol = X + 2 * half * NI + n0 + lr;

  v8f acc = {};
  #pragma unroll 4
  for (int k0 = 0; k0 < B_SZ; k0 += 4) {
    v2f a, b;
    a.x = acol[k0 * KO];
    a.y = acol[k0 * KO + KO];
    b.x = bcol[k0 * NI];
    b.y = bcol[k0 * NI + NI];
    acc = __builtin_amdgcn_wmma_f32_16x16x4_f32(false, a, false, b,
                                                (short)0, acc, false, false);
  }

  float* yrow = Y + (m0 + 8 * half) * NI + n0 + lr;
  #pragma unroll
  for (int r = 0; r < 8; ++r) yrow[r * NI] = acc[r];
}

// ---------------- host orchestration ----------------

extern "C" void kernel_launch(void* const* d_in, const int* in_sizes, int n_in,
                              void* d_out, int out_size, void* d_ws, size_t ws_size,
                              hipStream_t stream) {
  const float* X = (const float*)d_in[0];   // [256,1152,8]  -> X[b, n*8+i]
  const float* W = (const float*)d_in[1];   // [1,1152,10,16,8]
  float* out = (float*)d_out;               // [256,10,16,1] -> flat [256,160]
  float* ws  = (float*)d_ws;

  float* bij = ws;                          // 11,520
  float* c   = ws + 11520;                  // 11,520
  float* wct = ws + 23040;                  // 1,474,560
  float* s   = ws + 1497600;                // 40,960
  float* v   = ws + 1538560;                // 40,960
  float* y   = ws + 1579520;                // 1,474,560   (total ~12.2 MB)

  zero_kernel<<<(N_SZ * C_SZ + 255) / 256, 256, 0, stream>>>(bij, N_SZ * C_SZ);

  for (int it = 0; it < 3; ++it) {
    softmax_kernel<<<(N_SZ + 127) / 128, 128, 0, stream>>>(bij, c);
    make_wct_kernel<<<(NI * KO + 255) / 256, 256, 0, stream>>>(W, c, wct);
    gemm_s_kernel<<<20, 256, 0, stream>>>(X, wct, s);                  // 160 tiles / 8 waves
    squash_kernel<<<(B_SZ * KO + 255) / 256, 256, 0, stream>>>(
        s, v, (it == 2) ? out : (float*)nullptr);
    if (it < 2) {
      gemm_y_kernel<<<720, 256, 0, stream>>>(v, X, y);                 // 5760 tiles / 8 waves
      update_b_kernel<<<(N_SZ * C_SZ + 255) / 256, 256, 0, stream>>>(W, y, bij);
    }
  }
}